// AlibiPositionalBias_17660905521861
// MI455X (gfx1250) — compile-verified
//
#include <hip/hip_runtime.h>
#include <hip/hip_bf16.h>
#include <math.h>

// ALiBi positional bias + causal mask, fused elementwise over (H, S, S) fp32.
// Pure streaming: 512 MB traffic -> ~22 us floor at 23.3 TB/s HBM.
// Row-centric mapping: block owns rows (h,q); slope/q are row-uniform (SALU),
// per-element work is ~5 VALU. 128-bit NT loads/stores + global_prefetch_b8.

typedef float v4f __attribute__((ext_vector_type(4)));

__global__ void alibi_bias_rows(const float* __restrict__ scores,
                                float* __restrict__ out,
                                unsigned nrows,     // H * S
                                int logS)           // log2(S)
{
    const unsigned S          = 1u << logS;
    const unsigned tilesPerRow = S >> 2;            // float4 tiles per row
    const float    NEGINF     = -__builtin_inff();
    const float    SQRT_HALF  = 0.70710678118654752440f; // fp32(2^-0.5), exact rounding

    for (unsigned r = blockIdx.x; r < nrows; r += gridDim.x) {
        // ---- row-uniform setup (scalar-friendly) ----
        const unsigned q = r & (S - 1u);            // query position
        const unsigned h = r >> logS;               // head index
        // slope[h] = 2^(-(h+1)/2), exact: (odd ? sqrt(0.5) : 1) * 2^(-((h+1)>>1))
        const unsigned e     = h + 1u;
        const float    frac  = (e & 1u) ? SQRT_HALF : 1.0f;
        const float    slope = __builtin_ldexpf(frac, -(int)(e >> 1));
        const float    fq    = (float)(int)q;
        const size_t   rowBase = (size_t)r << logS; // r * S elements

        // ---- prefetch the next row this block will process (global_prefetch_b8) ----
        const unsigned rn = r + gridDim.x;
        if (rn < nrows) {
            const float* p = scores + ((size_t)rn << logS) + (size_t)threadIdx.x * 4u;
            __builtin_prefetch(p, 0, 1);
            __builtin_prefetch(p + (size_t)blockDim.x * 4u, 0, 1);
        }

        // ---- stream the row: each thread covers tilesPerRow/blockDim vec4 tiles ----
        for (unsigned t = threadIdx.x; t < tilesPerRow; t += blockDim.x) {
            const size_t off = rowBase + (size_t)t * 4u;
            v4f s = __builtin_nontemporal_load((const v4f*)(scores + off));

            const float fk = (float)(int)(t << 2) - fq;  // (k0 - q), exact in fp32
            v4f o;
#pragma unroll
            for (int j = 0; j < 4; ++j) {
                const float fkj  = fk + (float)j;        // exact integer add
                const float bias = fkj * slope;          // single rounding == reference
                const float res  = s[j] + bias;
                o[j] = (fkj > 0.0f) ? NEGINF : res;      // strict upper triangle -> -inf
            }
            __builtin_nontemporal_store(o, (v4f*)(out + off));
        }
    }
}

extern "C" void kernel_launch(void* const* d_in, const int* in_sizes, int n_in,
                              void* d_out, int out_size, void* d_ws, size_t ws_size,
                              hipStream_t stream) {
    const float* scores = (const float*)d_in[0];
    float*       out    = (float*)d_out;

    // out_size = H * S * S with H = 16, S a power of two (2048 in reference).
    const unsigned total  = (unsigned)out_size;
    const unsigned s2     = total / 16u;               // S*S
    const int      logS   = (31 - __builtin_clz(s2)) / 2;
    const unsigned S      = 1u << logS;
    const unsigned nrows  = total / S;                 // H * S

    const int threads = 256;                           // 8 wave32s per block
    unsigned blocks = nrows < 8192u ? nrows : 8192u;   // grid-stride: ~4 rows per block

    alibi_bias_rows<<<blocks, threads, 0, stream>>>(scores, out, nrows, logS);
}